// PGCN_twin_26388279066912
// MI455X (gfx1250) — compile-verified
//
#include <hip/hip_runtime.h>
#include <hip/hip_bf16.h>
#include <float.h>

// ---------------------------------------------------------------------------
// PGCN twin-tower GCN for MI455X (gfx1250, wave32, WMMA).
// GEMMs use V_WMMA_F32_16X16X4_F32 (full f32 precision), one wave = 16x32
// output tile (two independent accumulator chains, A fragment reused 2x).
// W panel staged in LDS with K-row-pair swizzle so each B fragment is one
// aligned 8-byte ds load (no repack moves in the WMMA loop).
// Scatter aggregation uses float4 gathers + global_atomic_add_f32.
// ---------------------------------------------------------------------------

typedef float v2f __attribute__((ext_vector_type(2)));
typedef float v8f __attribute__((ext_vector_type(8)));

#define NN 20000
#define EE 100000
#define GGR 64
#define FF 128

static inline int cdiv_h(long a, long b) { return (int)((a + b - 1) / b); }

__device__ __forceinline__ void atomicMaxF(float* addr, float v) {
  // Standard sign-split trick: signed-int max handles >=0, unsigned-int min handles <0.
  if (v >= 0.f) atomicMax((int*)addr, __float_as_int(v));
  else          atomicMin((unsigned int*)addr, __float_as_uint(v));
}

// ------------------------- degree / norm precompute -------------------------

__global__ void deg_init_k(float* deg4, float* deg1, int N) {
  int i = blockIdx.x * blockDim.x + threadIdx.x;
  if (i < N) {
    deg1[i] = 1.f;                       // self-loop weight 1
    #pragma unroll
    for (int k = 0; k < 4; ++k) deg4[k * N + i] = 1.f;
  }
}

__global__ void deg_edge_k(const int* __restrict__ ei, const float* __restrict__ ea,
                           float* deg4, float* deg1, int N, int E) {
  int e = blockIdx.x * blockDim.x + threadIdx.x;
  if (e < E) {
    int col = ei[E + e];                 // target
    atomicAdd(&deg1[col], 1.f);
    #pragma unroll
    for (int k = 0; k < 4; ++k) atomicAdd(&deg4[k * N + col], ea[e * 4 + k]);
  }
}

__global__ void deg_to_dis_k(float* deg4, float* deg1, int N) {
  int i = blockIdx.x * blockDim.x + threadIdx.x;
  if (i < N) {
    float d = deg1[i];
    deg1[i] = (d > 0.f) ? 1.f / sqrtf(d) : 0.f;
    #pragma unroll
    for (int k = 0; k < 4; ++k) {
      float dk = deg4[k * N + i];
      deg4[k * N + i] = (dk > 0.f) ? 1.f / sqrtf(dk) : 0.f;
    }
  }
}

__global__ void edge_norms_k(const int* __restrict__ ei, const float* __restrict__ ea,
                             const float* __restrict__ dis4, const float* __restrict__ dis1,
                             float* __restrict__ norm1, float* __restrict__ norm2,
                             int N, int E) {
  int e = blockIdx.x * blockDim.x + threadIdx.x;
  if (e < E) {
    int row = ei[e], col = ei[E + e];
    norm2[e] = dis1[row] * dis1[col];    // unit edge weight (layers 2,3)
    #pragma unroll
    for (int k = 0; k < 4; ++k)
      norm1[e * 4 + k] = dis4[k * N + row] * ea[e * 4 + k] * dis4[k * N + col];
  }
}

// ------------------------------- WMMA GEMM ---------------------------------
// C[M x Nw](+coff, ldc) = (reluA ? relu(A) : A)[M x K] * W[K x Nw]
// Block: 256 threads (8 waves). Each wave owns a 16x32 tile (two 16x16 WMMA
// accumulators sharing one A fragment); block covers 128 rows x 32 cols.
// LDS W panel swizzle: element (kr, nc) -> sW[((kr>>1)*32 + nc)*2 + (kr&1)],
// so the (K=2p, K=2p+1) pair for one column is contiguous: each B fragment is
// a single aligned v2f LDS load; b0/b1 are 128B apart (ds_load_2addr_b64).
// A-frag (16x4 f32): lanes0-15 M=row, v[0]=K(2*hi), v[1]=K(2*hi+1).
// C/D (16x16 f32): acc[v] -> row M = 8*hi + v, col N = lane&15.
#define GEMM_BM 128

template <bool RELU_A>
__global__ __launch_bounds__(256) void gemm_wmma_f32(
    const float* __restrict__ A, const float* __restrict__ W,
    float* __restrict__ C, int M, int K, int Nw, int ldc, int coff) {
  extern __shared__ float sW[];          // K x 32 panel, pair-swizzled (K*128 bytes)

  const int nBase = blockIdx.y * 32;
  for (int idx = threadIdx.x; idx < K * 32; idx += 256) {
    int kr = idx >> 5, nc = idx & 31;    // coalesced global read
    sW[((kr >> 1) * 32 + nc) * 2 + (kr & 1)] = W[kr * Nw + nBase + nc];
  }
  __syncthreads();

  const int lane = threadIdx.x & 31;
  const int wave = threadIdx.x >> 5;
  const int lo = lane & 15;
  const int hi = lane >> 4;
  const int mTile = blockIdx.x * GEMM_BM + wave * 16;

  int arow = mTile + lo;
  if (arow >= M) arow = M - 1;           // clamp; OOB results discarded at store
  const float* __restrict__ Ap = A + (size_t)arow * K + 2 * hi;

  v8f acc0 = {};
  v8f acc1 = {};
  #pragma unroll 4
  for (int k0 = 0; k0 < K; k0 += 4) {
    float ax = Ap[k0];
    float ay = Ap[k0 + 1];
    if (RELU_A) { ax = fmaxf(ax, 0.f); ay = fmaxf(ay, 0.f); }
    v2f a; a[0] = ax; a[1] = ay;
    // row-pair index for K rows (k0+2*hi, k0+2*hi+1)
    const float* bp = sW + (((k0 >> 1) + hi) * 32 + lo) * 2;
    v2f b0 = *(const v2f*)(bp);          // cols [nBase, nBase+16)
    v2f b1 = *(const v2f*)(bp + 32);     // cols [nBase+16, nBase+32)
    acc0 = __builtin_amdgcn_wmma_f32_16x16x4_f32(false, a, false, b0,
                                                 (short)0, acc0, false, false);
    acc1 = __builtin_amdgcn_wmma_f32_16x16x4_f32(false, a, false, b1,
                                                 (short)0, acc1, false, false);
  }

  const int r0 = mTile + 8 * hi;
  #pragma unroll
  for (int v = 0; v < 8; ++v) {
    int r = r0 + v;
    if (r < M) {
      float* crow = C + (size_t)r * ldc + coff + nBase + lo;
      crow[0]  = acc0[v];
      crow[16] = acc1[v];
    }
  }
}

// --------------------------- aggregation (layer 1) --------------------------
// h1 is [N,512] = concat of the 4 conv outputs; channel k = cols [128k,128k+128).

__global__ void agg_init1_k(const float* __restrict__ h1, const float* __restrict__ dis4,
                            const float* __restrict__ bA, const float* __restrict__ bB,
                            const float* __restrict__ bC, const float* __restrict__ bD,
                            float* __restrict__ agg, int N) {
  int t = blockIdx.x * blockDim.x + threadIdx.x;   // N*128 threads, float4 each
  int i = t >> 7, q = t & 127;
  if (i < N) {
    int f0 = q * 4;
    int k = f0 >> 7;
    float ds = dis4[k * N + i];
    float s = ds * ds;                               // self-loop norm
    const float* b = (k == 0) ? bA : (k == 1) ? bB : (k == 2) ? bC : bD;
    int fk = f0 & 127;
    float4 h = *(const float4*)(h1 + (size_t)i * 512 + f0);
    float4 o;
    o.x = s * h.x + b[fk + 0];
    o.y = s * h.y + b[fk + 1];
    o.z = s * h.z + b[fk + 2];
    o.w = s * h.w + b[fk + 3];
    *(float4*)(agg + (size_t)i * 512 + f0) = o;
  }
}

__global__ void agg_edge1_k(const int* __restrict__ ei, const float* __restrict__ norm1,
                            const float* __restrict__ h1, float* __restrict__ agg, int E) {
  int t = blockIdx.x * blockDim.x + threadIdx.x;   // E*128 threads
  int e = t >> 7, q = t & 127;
  if (e < E) {
    int f0 = q * 4;
    int k = f0 >> 7;
    float nm = norm1[e * 4 + k];
    if (nm != 0.f) {
      int row = ei[e], col = ei[E + e];
      float4 h = *(const float4*)(h1 + (size_t)row * 512 + f0);
      float* dst = agg + (size_t)col * 512 + f0;
      atomicAdd(dst + 0, nm * h.x);
      atomicAdd(dst + 1, nm * h.y);
      atomicAdd(dst + 2, nm * h.z);
      atomicAdd(dst + 3, nm * h.w);
    }
  }
}

// ------------------------- aggregation (layers 2/3) -------------------------

__global__ void agg_init_k(const float* __restrict__ h, const float* __restrict__ dis1,
                           const float* __restrict__ bias, float* __restrict__ agg,
                           int N, int Fw) {
  int t = blockIdx.x * blockDim.x + threadIdx.x;   // N*(Fw/4)
  int q4 = Fw >> 2;
  int i = t / q4, q = t % q4;
  if (i < N) {
    int f0 = q * 4;
    float ds = dis1[i];
    float s = ds * ds;
    float4 hv = *(const float4*)(h + (size_t)i * Fw + f0);
    float4 o;
    o.x = s * hv.x + bias[f0 + 0];
    o.y = s * hv.y + bias[f0 + 1];
    o.z = s * hv.z + bias[f0 + 2];
    o.w = s * hv.w + bias[f0 + 3];
    *(float4*)(agg + (size_t)i * Fw + f0) = o;
  }
}

__global__ void agg_edge_k(const int* __restrict__ ei, const float* __restrict__ norm2,
                           const float* __restrict__ h, float* __restrict__ agg,
                           int E, int Fw) {
  int t = blockIdx.x * blockDim.x + threadIdx.x;   // E*(Fw/4)
  int q4 = Fw >> 2;
  int e = t / q4, q = t % q4;
  if (e < E) {
    float nm = norm2[e];
    if (nm != 0.f) {
      int f0 = q * 4;
      int row = ei[e], col = ei[E + e];
      float4 hv = *(const float4*)(h + (size_t)row * Fw + f0);
      float* dst = agg + (size_t)col * Fw + f0;
      atomicAdd(dst + 0, nm * hv.x);
      atomicAdd(dst + 1, nm * hv.y);
      atomicAdd(dst + 2, nm * hv.z);
      atomicAdd(dst + 3, nm * hv.w);
    }
  }
}

// --------------------------------- pooling ---------------------------------

__global__ void pool_clear_k(float* psum, float* pmax, float* pcnt) {
  int t = blockIdx.x * blockDim.x + threadIdx.x;   // GGR*FF
  if (t < GGR * FF) {
    psum[t] = 0.f;
    pmax[t] = -FLT_MAX;
    if (t < GGR) pcnt[t] = 0.f;
  }
}

__global__ void pool_scatter_k(const float* __restrict__ agg3, const int* __restrict__ batch,
                               float* psum, float* pmax, float* pcnt, int N) {
  int t = blockIdx.x * blockDim.x + threadIdx.x;   // N*FF
  int i = t >> 7, f = t & 127;
  if (i < N) {
    int g = batch[i];
    float v = agg3[(size_t)i * FF + f];
    atomicAdd(&psum[g * FF + f], v);
    atomicMaxF(&pmax[g * FF + f], v);
    if (f == 0) atomicAdd(&pcnt[g], 1.f);
  }
}

__global__ void pool_final_k(const float* psum, const float* pmax, const float* pcnt,
                             float* pool) {              // pool: [GGR][2*FF]
  int t = blockIdx.x * blockDim.x + threadIdx.x;         // GGR*256
  if (t < GGR * 2 * FF) {
    int g = t >> 8, f = t & 255;
    float c = pcnt[g];
    float o = (f < FF) ? psum[g * FF + f] / fmaxf(c, 1.f)
                       : ((c > 0.f) ? pmax[g * FF + (f - FF)] : 0.f);
    pool[t] = o;
  }
}

// ---------------------------------- head -----------------------------------

__global__ __launch_bounds__(512) void head_k(const float* __restrict__ pool0,
                                              const float* __restrict__ pool1,
                                              const float* __restrict__ Wm1,
                                              const float* __restrict__ bm1,
                                              const float* __restrict__ Wm2,
                                              const float* __restrict__ bm2,
                                              float* __restrict__ out) {
  __shared__ float sh[GGR * 8];
  int t = threadIdx.x;                               // 512 threads = 64 graphs x 8 units
  int g = t >> 3, j = t & 7;
  float s = bm1[j];
  for (int k = 0; k < 512; ++k) {
    float p = (k < 256) ? pool0[g * 256 + k] : pool1[g * 256 + (k - 256)];
    s += p * Wm1[k * 8 + j];
  }
  sh[g * 8 + j] = fmaxf(s, 0.f);
  __syncthreads();
  if (t < GGR * 2) {
    int gg = t >> 1, c = t & 1;
    float o = bm2[c];
    #pragma unroll
    for (int k = 0; k < 8; ++k) o += sh[gg * 8 + k] * Wm2[k * 2 + c];
    out[gg * 2 + c] = o;
  }
}

// --------------------------------- launch ----------------------------------

extern "C" void kernel_launch(void* const* d_in, const int* in_sizes, int n_in,
                              void* d_out, int out_size, void* d_ws, size_t ws_size,
                              hipStream_t stream) {
  const int N = NN, E = EE, G = GGR;

  const float* W1[4] = {(const float*)d_in[8],  (const float*)d_in[10],
                        (const float*)d_in[12], (const float*)d_in[14]};
  const float* b1[4] = {(const float*)d_in[9],  (const float*)d_in[11],
                        (const float*)d_in[13], (const float*)d_in[15]};
  const float* W2  = (const float*)d_in[16];
  const float* b2  = (const float*)d_in[17];
  const float* W3  = (const float*)d_in[18];
  const float* b3  = (const float*)d_in[19];
  const float* Wm1 = (const float*)d_in[20];
  const float* bm1 = (const float*)d_in[21];
  const float* Wm2 = (const float*)d_in[22];
  const float* bm2 = (const float*)d_in[23];

  // Workspace carve-up (floats). Peak ~85 MB.
  float* ws = (float*)d_ws;
  size_t o = 0;
  float* dis4  = ws + o; o += (size_t)4 * N;
  float* dis1  = ws + o; o += (size_t)N;
  float* norm1 = ws + o; o += (size_t)4 * E;
  float* norm2 = ws + o; o += (size_t)E;
  float* bufA  = ws + o; o += (size_t)N * 512;   // h1 -> h2 -> h3
  float* bufB  = ws + o; o += (size_t)N * 512;   // agg1 -> agg2 -> agg3
  float* psum  = ws + o; o += (size_t)G * FF;
  float* pmax  = ws + o; o += (size_t)G * FF;
  float* pcnt  = ws + o; o += (size_t)G;
  float* pools = ws + o; o += (size_t)2 * G * 256;

  const int B = 256;
  const int gx = cdiv_h(N, GEMM_BM);               // 157 row-tiles of 128

  for (int t = 0; t < 2; ++t) {
    const float* x     = (const float*)d_in[t == 0 ? 0 : 3];
    const float* ea    = (const float*)d_in[t == 0 ? 1 : 4];
    const int*   ei    = (const int*)  d_in[t == 0 ? 2 : 5];
    const int*   batch = (const int*)  d_in[6 + t];
    float* pool = pools + (size_t)t * G * 256;

    // degrees -> dis -> per-edge norms
    deg_init_k<<<cdiv_h(N, B), B, 0, stream>>>(dis4, dis1, N);
    deg_edge_k<<<cdiv_h(E, B), B, 0, stream>>>(ei, ea, dis4, dis1, N, E);
    deg_to_dis_k<<<cdiv_h(N, B), B, 0, stream>>>(dis4, dis1, N);
    edge_norms_k<<<cdiv_h(E, B), B, 0, stream>>>(ei, ea, dis4, dis1, norm1, norm2, N, E);

    // Layer 1: h1 = x @ [W1A|W1B|W1C|W1D]  (4 column-block GEMMs into [N,512])
    for (int k = 0; k < 4; ++k)
      gemm_wmma_f32<false><<<dim3(gx, 128 / 32), B, (size_t)128 * 32 * 4, stream>>>(
          x, W1[k], bufA, N, 128, 128, 512, k * 128);
    agg_init1_k<<<cdiv_h((size_t)N * 128, B), B, 0, stream>>>(bufA, dis4,
                                                              b1[0], b1[1], b1[2], b1[3],
                                                              bufB, N);
    agg_edge1_k<<<cdiv_h((size_t)E * 128, B), B, 0, stream>>>(ei, norm1, bufA, bufB, E);

    // Layer 2: h2 = relu(agg1) @ W2   (relu folded into A-fragment load)
    gemm_wmma_f32<true><<<dim3(gx, 256 / 32), B, (size_t)512 * 32 * 4, stream>>>(
        bufB, W2, bufA, N, 512, 256, 256, 0);
    agg_init_k<<<cdiv_h((size_t)N * 64, B), B, 0, stream>>>(bufA, dis1, b2, bufB, N, 256);
    agg_edge_k<<<cdiv_h((size_t)E * 64, B), B, 0, stream>>>(ei, norm2, bufA, bufB, E, 256);

    // Layer 3: h3 = relu(agg2) @ W3
    gemm_wmma_f32<true><<<dim3(gx, 128 / 32), B, (size_t)256 * 32 * 4, stream>>>(
        bufB, W3, bufA, N, 256, 128, 128, 0);
    agg_init_k<<<cdiv_h((size_t)N * 32, B), B, 0, stream>>>(bufA, dis1, b3, bufB, N, 128);
    agg_edge_k<<<cdiv_h((size_t)E * 32, B), B, 0, stream>>>(ei, norm2, bufA, bufB, E, 128);

    // Pool (mean || max)
    pool_clear_k<<<cdiv_h(G * FF, B), B, 0, stream>>>(psum, pmax, pcnt);
    pool_scatter_k<<<cdiv_h((size_t)N * FF, B), B, 0, stream>>>(bufB, batch,
                                                                psum, pmax, pcnt, N);
    pool_final_k<<<cdiv_h(G * 256, B), B, 0, stream>>>(psum, pmax, pcnt, pool);
  }

  // Head MLP: [64,512] -> relu(8) -> 2
  head_k<<<1, 512, 0, stream>>>(pools, pools + (size_t)G * 256,
                                Wm1, bm1, Wm2, bm2, (float*)d_out);
}